// CostVolumeLayer_35476429865604
// MI455X (gfx1250) — compile-verified
//
#include <hip/hip_runtime.h>

typedef float v2f __attribute__((ext_vector_type(2)));
typedef float v8f __attribute__((ext_vector_type(8)));

#define BATCH 8
#define CH    64
#define HGT   128
#define WID   256
#define SR    4
#define ND    81      // (2*SR+1)^2
#define TW    144     // staged tgt row width: 128 + 2*8 halo

__device__ __forceinline__ int iabs(int x) { return x < 0 ? -x : x; }

// Index of displacement (di,dj) in the reference's stacking order.
__device__ __forceinline__ int dindex(int di, int dj) {
  if (di == 0 && dj == 0) return 0;
  if (dj == 0) return 1 + (iabs(di) - 1) * 20 + ((di > 0) ? 0 : 1);
  if (di == 0) return 1 + (iabs(dj) - 1) * 20 + ((dj > 0) ? 2 : 3);
  return 1 + (iabs(di) - 1) * 20 + 4 + (iabs(dj) - 1) * 4
           + ((di > 0) ? 0 : 1) + ((dj > 0) ? 0 : 2);
}

__global__ __launch_bounds__(256) void
costvol_wmma_f32(const float* __restrict__ src,
                 const float* __restrict__ tgt,
                 float* __restrict__ out) {
  const int Wc   = blockIdx.x;   // 0..1   : 128-pixel chunk of the row
  const int h    = blockIdx.y;   // 0..127
  const int bb   = blockIdx.z;   // 0..7
  const int tid  = threadIdx.x;
  const int wid  = tid >> 5;     // wave id 0..7 -> 16-pixel group
  const int lane = tid & 31;
  const int n    = lane & 15;    // N index (and M index for A loads)
  const int hf   = lane >> 4;    // lane-half: selects K pair / M half

  const int W0 = Wc * 128;
  const int w0 = W0 + wid * 16;

  // tgt row, pair-interleaved: (c,p) -> dword ((c>>1)*TW + p)*2 + (c&1)
  // => channels (c, c+1) at one pixel are one aligned 8-byte ds_load_b64.
  __shared__ float lt[CH * TW];          // 36 KB
  __shared__ int   dtab[ND];             // displacement-order lookup

  if (tid < ND) dtab[tid] = dindex(tid / 9 - 4, tid % 9 - 4);

  // ---- A fragments: src row h, this wave's 16 pixels, all 64 channels ----
  // ISA 16x4 f32 A layout: lanes 0-15 hold K = 4k+{0,1}, lanes 16-31 K = 4k+{2,3}.
  float ax[16], ay[16];
  {
    const float* sp = src + (size_t)bb * CH * HGT * WID
                          + (size_t)h * WID + (w0 + n);
#pragma unroll
    for (int k = 0; k < 16; ++k) {
      const int c = 4 * k + 2 * hf;
      ax[k] = sp[(size_t)c       * HGT * WID];
      ay[k] = sp[(size_t)(c + 1) * HGT * WID];
    }
  }

  // ---- per-lane scatter slots, hoisted out of the row loop ----
  // slot s = t*8 + r : value D_t[r] = G[m, n] with m = r + 8*hf,
  // tgt column offset sj = n - m - 8 + 16*t, reference dj = -sj.
  int  wofs[16];   // output column  w0 + m
  int  tofs[16];   // (dj+4), clamped; dj-part of dtab index
  bool val [16];
#pragma unroll
  for (int t = 0; t < 2; ++t)
#pragma unroll
    for (int r = 0; r < 8; ++r) {
      const int s  = t * 8 + r;
      const int m  = r + 8 * hf;
      const int sj = n - m - 8 + 16 * t;
      const int dj = -sj;
      val[s]  = (sj >= -SR) && (sj <= SR);
      int u   = dj + 4;
      u       = u < 0 ? 0 : (u > 8 ? 8 : u);  // clamp (unused when invalid)
      tofs[s] = u;
      wofs[s] = w0 + m;
    }

  // out + (b, :, h, :) base;  offset = dIdx*(HGT*WID) + w,  HGT*WID = 1<<15
  float* outP = out + (size_t)bb * ND * HGT * WID + (size_t)h * WID;

  const int bofs = (hf * TW + wid * 16 + n) * 2;   // LDS dword base of B frags

#pragma unroll
  for (int si = -SR; si <= SR; ++si) {
    // ---- stage tgt row h+si (zero-padded halo), float4 granules ----
    const int  hh    = h + si;
    const bool rowok = (hh >= 0) && (hh < HGT);
#pragma unroll
    for (int i = 0; i < 9; ++i) {                  // 2304 float4 / 256 threads
      const int idx4 = tid + i * 256;
      const int c = idx4 / 36;
      const int q = idx4 - c * 36;
      const int w = W0 - 8 + 4 * q;
      float4 v = make_float4(0.f, 0.f, 0.f, 0.f);
      if (rowok && (unsigned)w < 253u)             // all-4-or-nothing (4-aligned)
        v = *(const float4*)&tgt[(((size_t)bb * CH + c) * HGT + hh) * WID + w];
      const int d = ((c >> 1) * TW + 4 * q) * 2 + (c & 1);
      lt[d]     = v.x;
      lt[d + 2] = v.y;
      lt[d + 4] = v.z;
      lt[d + 6] = v.w;
    }
    __syncthreads();

    // ---- two 16x16 Gram tiles (tgt offsets -8 and +8), K = 64 ----
    v8f acc0 = {0.f,0.f,0.f,0.f,0.f,0.f,0.f,0.f};
    v8f acc1 = {0.f,0.f,0.f,0.f,0.f,0.f,0.f,0.f};
#pragma unroll
    for (int k = 0; k < 16; ++k) {
      v2f a  = { ax[k], ay[k] };
      v2f b0 = *(const v2f*)(lt + bofs + k * 2 * TW);        // ds_load_b64
      v2f b1 = *(const v2f*)(lt + bofs + k * 2 * TW + 32);   // tile1: +16 px
      acc0 = __builtin_amdgcn_wmma_f32_16x16x4_f32(
          false, a, false, b0, (short)0, acc0, false, false);
      acc1 = __builtin_amdgcn_wmma_f32_16x16x4_f32(
          false, a, false, b1, (short)0, acc1, false, false);
    }
    __syncthreads();   // LDS reads done before next row is staged

    // ---- scatter the 9 valid diagonals of each tile ----
    const int tb = (-si + 4) * 9;                  // (di+4)*9, compile-time
#pragma unroll
    for (int r = 0; r < 8; ++r) {
      if (val[r]) {
        const unsigned off = ((unsigned)dtab[tb + tofs[r]] << 15) + wofs[r];
        outP[off] = acc0[r];
      }
      if (val[8 + r]) {
        const unsigned off = ((unsigned)dtab[tb + tofs[8 + r]] << 15) + wofs[8 + r];
        outP[off] = acc1[r];
      }
    }
  }
}

extern "C" void kernel_launch(void* const* d_in, const int* in_sizes, int n_in,
                              void* d_out, int out_size, void* d_ws, size_t ws_size,
                              hipStream_t stream) {
  (void)in_sizes; (void)n_in; (void)d_ws; (void)ws_size; (void)out_size;
  const float* src = (const float*)d_in[0];
  const float* tgt = (const float*)d_in[1];
  float* out = (float*)d_out;
  dim3 grid(WID / 128, HGT, BATCH);
  costvol_wmma_f32<<<grid, 256, 0, stream>>>(src, tgt, out);
}